// PlasticFCNetwork_63599875719445
// MI455X (gfx1250) — compile-verified
//
#include <hip/hip_runtime.h>

// ============================================================================
// Plastic (Hebbian) 2-layer RNN for MI455X (gfx1250).
//
// 16 persistent workgroups x 256 threads; WG w owns output-column slice
// E_w = [16w, 16w+16) for every batch/layer. Hebb state (8 MB) lives in d_ws
// (L2-resident, per-WG-sliced). The batched matmul Y(16x256) @ W[:,E_w]
// (256x16) maps onto V_WMMA_F32_16X16X4_F32 with K split across 8 waves.
//
// Hebb update is applied LAZILY: at step t we fuse the deferred update from
// step t-1 (h <- (1-eta)h + eta*yin_{t-1} (x) yin_t) into the plastic-read
// pass, so hebb is touched once-read + once-write per step instead of 3x.
// Y state is a ring of 3 global buffers per layer so cur/prev/write never
// alias within a phase. Grid-wide sync (atomic arrive + generation)
// separates the 256 serial layer-steps; an init kernel re-zeroes all
// persistent scratch every call so graph replay is deterministic.
// ============================================================================

typedef __attribute__((ext_vector_type(2))) float v2f;
typedef __attribute__((ext_vector_type(4))) float v4f;
typedef __attribute__((ext_vector_type(8))) float v8f;

#define Bb 16
#define Tt 128
#define Dd 256
#define Llay 2
#define NWG 16           // one WG per 16-column slice of D=256
#define NTHREADS 256     // 16 batches x 16 slice-columns; 8 wave32s
#define NWAVES 8
#define YPITCH 257       // LDS pitch (bank-conflict-free WMMA A loads)

// ---- d_ws layout (bytes) ----
#define WS_SYNC   0                          // 2 unsigned (arrive, gen)
#define WS_YG     256                        // Y ring: 2 lay * 3 buf * 16*256 f32 = 96 KB
#define WS_HEBB   (256 + Llay * 3 * Bb * Dd * 4)          // = 98560 (16B aligned)
#define WS_TOTAL  (WS_HEBB + (size_t)Llay * NWG * Bb * Dd * 16 * 4)

__global__ void pfc_init_zero(float* ws, unsigned n) {
    unsigned i = blockIdx.x * blockDim.x + threadIdx.x;
    unsigned stride = gridDim.x * blockDim.x;
    for (; i < n; i += stride) ws[i] = 0.0f;
}

__device__ __forceinline__ void grid_sync(unsigned* arrive, unsigned* gen,
                                          unsigned& localGen) {
    __threadfence();
    __syncthreads();
    if (threadIdx.x == 0) {
        const unsigned target = ++localGen;
        if (atomicAdd(arrive, 1u) == NWG - 1) {
            __hip_atomic_store(arrive, 0u, __ATOMIC_RELAXED, __HIP_MEMORY_SCOPE_AGENT);
            __hip_atomic_fetch_add(gen, 1u, __ATOMIC_RELEASE, __HIP_MEMORY_SCOPE_AGENT);
        } else {
            while (__hip_atomic_load(gen, __ATOMIC_ACQUIRE, __HIP_MEMORY_SCOPE_AGENT) < target)
                __builtin_amdgcn_s_sleep(2);
        }
    }
    __syncthreads();
}

__global__ __launch_bounds__(NTHREADS, 1)
void pfc_persistent(const int* __restrict__ x, const float* __restrict__ emb,
                    const float* __restrict__ wmat, const float* __restrict__ alpha,
                    const float* __restrict__ etas, float* __restrict__ out,
                    float* __restrict__ wsf, unsigned* __restrict__ sync) {
    __shared__ float Yc[Bb * YPITCH];        // yin_t   (16x256, pitch 257)
    __shared__ float Yp[Bb * YPITCH];        // yin_{t-1}
    __shared__ float GP[NWAVES * 256];       // per-wave WMMA partial C tiles
    __shared__ float PP[256 * 4];            // plastic partials [b][e][dchunk]
    __shared__ float PS[NTHREADS];           // softmax partials
    __shared__ float RS[Bb];                 // softmax row sums

    const int tid  = threadIdx.x;
    const int wg   = blockIdx.x;             // slice id: columns [16*wg, 16*wg+16)
    const int b    = tid >> 4;               // batch row (ownership mapping)
    const int et   = tid & 15;               // column within slice
    const int col  = wg * 16 + et;
    const int lane = tid & 31;
    const int wave = tid >> 5;               // 0..7
    const int half = lane >> 4;              // WMMA half-lane select
    const int lm   = lane & 15;
    // fused-loop mapping: (b, d-chunk, e-quad)
    const int fdc  = (tid >> 2) & 3;         // d chunk: 64 consecutive d
    const int feg  = tid & 3;                // e quad within slice
    const int colq = wg * 16 + feg * 4;

    float*    Yg     = wsf + WS_YG / 4;      // [lay][ring3][b][d]
    float*    hebb   = wsf + WS_HEBB / 4;    // [lay][wg][b][d][16]
    unsigned* arrive = sync;
    unsigned* gen    = sync + 1;
    unsigned  localGen = 0;

    const float eta0 = etas[0], eta1 = etas[1];
    float inp_carry = 0.0f;

    for (int t = 0; t < Tt; ++t) {
        const int rc = t % 3;                // ring slot holding yin_t
        const int rp = (t + 2) % 3;          // ring slot holding yin_{t-1}
        const int rw = (t + 1) % 3;          // ring slot to write yout_t

        for (int l = 0; l < Llay; ++l) {
            // ---- stage yin_t and yin_{t-1} into LDS (vectorized) ----
            const float* ycsrc = Yg + ((l * 3 + rc) * Bb) * Dd;
            const float* ypsrc = Yg + ((l * 3 + rp) * Bb) * Dd;
            for (int i4 = tid; i4 < Bb * Dd / 4; i4 += NTHREADS) {
                const int bb = i4 >> 6, dd = (i4 & 63) * 4;
                v4f vc = *(const v4f*)(ycsrc + i4 * 4);
                v4f vp = *(const v4f*)(ypsrc + i4 * 4);
                float* dc_ = Yc + bb * YPITCH + dd;
                float* dp_ = Yp + bb * YPITCH + dd;
                dc_[0] = vc.x; dc_[1] = vc.y; dc_[2] = vc.z; dc_[3] = vc.w;
                dp_[0] = vp.x; dp_[1] = vp.y; dp_[2] = vp.z; dp_[3] = vp.w;
            }
            __syncthreads();

            // ---- GEMM: C(16x16) = Y(16x256) @ W[:,E_w](256x16) via f32 WMMA.
            // K=256 split across 8 waves; B operands stream from L2.
            v8f acc = {};
            const float* gw = wmat + l * Dd * Dd + wg * 16 + lm;
            #pragma unroll
            for (int kk = 0; kk < 32; kk += 4) {
                const int k  = wave * 32 + kk;
                const int ka = k + half * 2;         // half-lane K split
                v2f a, bv;
                a.x  = Yc[lm * YPITCH + ka];
                a.y  = Yc[lm * YPITCH + ka + 1];
                bv.x = gw[ka * Dd];
                bv.y = gw[(ka + 1) * Dd];
                acc = __builtin_amdgcn_wmma_f32_16x16x4_f32(
                    false, a, false, bv, (short)0, acc, false, false);
            }
            #pragma unroll
            for (int j = 0; j < 8; ++j)              // C layout: M = j + 8*half
                GP[wave * 256 + (j + half * 8) * 16 + lm] = acc[j];

            // ---- fused pass: apply deferred hebb update, write back, and
            //      accumulate plastic with the UPDATED hebb.
            //      h_new[d,e] = (1-eta)*h + eta*Yp[b,d]*Yc[b,e]
            //      plastic[b,e] += Yc[b,d] * alpha[d,e] * h_new[d,e]
            const float eta = (l == 0) ? eta0 : eta1;
            const float om  = 1.0f - eta;
            float* hb4 = hebb + (((size_t)(l * NWG + wg) * Bb + b) * Dd + fdc * 64) * 16
                              + feg * 4;
            const float* al4 = alpha + l * Dd * Dd + colq + (fdc * 64) * Dd;
            v4f yce;                                 // Yc[b, colq..colq+3]
            yce.x = Yc[b * YPITCH + colq + 0];
            yce.y = Yc[b * YPITCH + colq + 1];
            yce.z = Yc[b * YPITCH + colq + 2];
            yce.w = Yc[b * YPITCH + colq + 3];
            const v4f etaYc = eta * yce;
            v4f pl = {};
            __builtin_prefetch(hb4, 1, 1);           // global_prefetch_b8
            #pragma unroll 4
            for (int i = 0; i < 64; ++i) {
                v4f h  = *(v4f*)(hb4 + i * 16);
                v4f a4 = *(const v4f*)(al4 + i * Dd);
                const float ypd = Yp[b * YPITCH + fdc * 64 + i];
                const float ycd = Yc[b * YPITCH + fdc * 64 + i];
                h = om * h + etaYc * ypd;
                *(v4f*)(hb4 + i * 16) = h;
                pl += ycd * (a4 * h);
            }
            PP[(b * 16 + feg * 4 + 0) * 4 + fdc] = pl.x;
            PP[(b * 16 + feg * 4 + 1) * 4 + fdc] = pl.y;
            PP[(b * 16 + feg * 4 + 2) * 4 + fdc] = pl.z;
            PP[(b * 16 + feg * 4 + 3) * 4 + fdc] = pl.w;
            __syncthreads();

            // ---- combine: GEMM + plastic + input, then activation ----
            float g = 0.0f;
            #pragma unroll
            for (int wv = 0; wv < NWAVES; ++wv) g += GP[wv * 256 + tid];
            float plastic = 0.0f;
            #pragma unroll
            for (int dc2 = 0; dc2 < 4; ++dc2) plastic += PP[tid * 4 + dc2];

            float inp;
            if (l == 0) {
                const int tok = x[b * Tt + t];
                inp = emb[tok * Dd + col];
            } else {
                inp = inp_carry;                     // layer-0 yout at same (b,col)
            }
            const float yout = tanhf(g + plastic + inp);
            inp_carry = yout;

            // ---- publish new Y slice into ring slot rw ----
            Yg[((l * 3 + rw) * Bb + b) * Dd + col] = yout;

            grid_sync(arrive, gen, localGen);
        }

        // ---- output: o[b,t,:] = softmax(sigmoid(y1[b,:])) ----
        const float* y1 = Yg + ((1 * 3 + rw) * Bb) * Dd;   // layer-1 yout_t
        for (int i4 = tid; i4 < Bb * Dd / 4; i4 += NTHREADS) {
            const int bb = i4 >> 6, dd = (i4 & 63) * 4;
            v4f v = *(const v4f*)(y1 + i4 * 4);
            float* dst = Yc + bb * YPITCH + dd;
            dst[0] = v.x; dst[1] = v.y; dst[2] = v.z; dst[3] = v.w;
        }
        __syncthreads();

        float ps = 0.0f;
        #pragma unroll
        for (int i = 0; i < 16; ++i) {
            const float yv = Yc[b * YPITCH + et * 16 + i];
            const float s  = 1.0f / (1.0f + expf(-yv));
            ps += expf(s);
        }
        PS[tid] = ps;
        __syncthreads();
        if (et == 0) {
            float s = 0.0f;
            #pragma unroll
            for (int i = 0; i < 16; ++i) s += PS[b * 16 + i];
            RS[b] = s;
        }
        __syncthreads();
        {
            const float yv = Yc[b * YPITCH + col];
            const float s  = 1.0f / (1.0f + expf(-yv));
            out[(b * Tt + t) * Dd + col] = expf(s) / RS[b];
        }
        __syncthreads();   // protect Yc before next step restages it
    }
}

extern "C" void kernel_launch(void* const* d_in, const int* in_sizes, int n_in,
                              void* d_out, int out_size, void* d_ws, size_t ws_size,
                              hipStream_t stream) {
    const int*   x     = (const int*)d_in[0];
    const float* emb   = (const float*)d_in[1];
    const float* wmat  = (const float*)d_in[2];
    const float* alpha = (const float*)d_in[3];
    const float* etas  = (const float*)d_in[4];
    float*       out   = (float*)d_out;
    float*       wsf   = (float*)d_ws;
    unsigned*    sync  = (unsigned*)d_ws;

    // Zero all persistent scratch (sync vars, Y ring, hebb state) every call
    // so graph replays are deterministic.
    const unsigned nzero = (unsigned)(WS_TOTAL / 4);
    pfc_init_zero<<<512, 256, 0, stream>>>(wsf, nzero);

    pfc_persistent<<<NWG, NTHREADS, 0, stream>>>(x, emb, wmat, alpha, etas,
                                                 out, wsf, sync);
    (void)in_sizes; (void)n_in; (void)out_size; (void)ws_size;
}